// CircuitGNN_89833535963611
// MI455X (gfx1250) — compile-verified
//
#include <hip/hip_runtime.h>

typedef __attribute__((ext_vector_type(2))) float v2f;
typedef __attribute__((ext_vector_type(8))) float v8f;
typedef __attribute__((ext_vector_type(4))) unsigned int v4u;
typedef __attribute__((ext_vector_type(8))) int v8i;
typedef __attribute__((ext_vector_type(4))) int v4i;

#define HID 128

// ---------------------------------------------------------------- utilities
__global__ void k_zero(float* __restrict__ p, long long n) {
  long long i = (long long)blockIdx.x * blockDim.x + threadIdx.x;
  const long long stride = (long long)gridDim.x * blockDim.x;
  for (; i < n; i += stride) p[i] = 0.0f;
}

__global__ void k_degree(const int* __restrict__ dst, float* __restrict__ deg, int nE) {
  const int e = blockIdx.x * blockDim.x + threadIdx.x;
  if (e < nE) atomicAdd(&deg[dst[e]], 1.0f);
}

__global__ void k_invdeg(float* __restrict__ deg, int n) {
  const int i = blockIdx.x * blockDim.x + threadIdx.x;
  if (i < n) deg[i] = 1.0f / fmaxf(deg[i], 1.0f);
}

// ------------------------------------------------ TDM 2-D tile -> LDS stage
// Loads a [16 x K] f32 tile starting at gptr (row stride K elements) into LDS
// at byte offset lds_off, padding each row by 4 dwords (LDS stride K+4).
// D# per CDNA5 ISA: group0 {count=1, lds_addr, global_addr, type=2},
// group1 {data_size=4B, pad_enable, pad_interval, pad_amount=4dw,
//         tensor_dim0=K, tensor_dim1=16, tile_dim0=K, tile_dim1=16, stride0=K},
// higher groups zero (2-D tensor: tile_dim2/3/4 unused).
template <int K>
__device__ __forceinline__ void tdm_load_tile16(const float* gptr, unsigned lds_off) {
  constexpr unsigned padInterval = (K == 128) ? 6u : 5u;  // 128 / 64 dwords
  constexpr unsigned padAmount = 3u;                      // 4 dwords
  const unsigned long long ga = (unsigned long long)gptr;
  v4u g0;
  g0[0] = 1u;                                   // count=1, user descriptor
  g0[1] = lds_off;                              // LDS byte address
  g0[2] = (unsigned)ga;                         // global_addr[31:0]
  g0[3] = (unsigned)((ga >> 32) & 0x1FFFFFFu) | (2u << 30);  // addr[56:32]|type=2
  v8i g1;
  g1[0] = (int)((2u << 16) | (1u << 20) | (padInterval << 22) | (padAmount << 25));
  g1[1] = (int)(((unsigned)K & 0xFFFFu) << 16);                  // tensor_dim0 lo
  g1[2] = (int)((((unsigned)K >> 16) & 0xFFFFu) | (16u << 16));  // dim0 hi | dim1 lo
  g1[3] = (int)(((unsigned)K & 0xFFFFu) << 16);                  // dim1 hi=0 | tile_dim0=K
  g1[4] = 16;                                                    // tile_dim1=16, tile_dim2=0
  g1[5] = K;                                                     // tensor_dim0_stride lo
  g1[6] = 0;                                                     // stride0 hi | stride1 lo
  g1[7] = 0;                                                     // stride1 hi
  const v4i z4 = {0, 0, 0, 0};
  const v8i z8 = {0, 0, 0, 0, 0, 0, 0, 0};
  // clang-23 / therock-10.0 form: (g0, g1, g2, g3, g4, cpol)
  __builtin_amdgcn_tensor_load_to_lds(g0, g1, z4, z4, z8, 0);
}

// ------------------------------------------------------------ WMMA f32 GEMM
// C[M x 128] = (A (+ Agg*invdeg)) [M x K] @ W[K x 128] + bias[128]
// 256 threads = 8 waves; block covers 16 rows x 128 cols. A strip (and Agg
// strip when FUSE) staged into LDS by the TDM; each wave runs a branch-free
// K-loop of V_WMMA_F32_16X16X4_F32 with ds-sourced A fragments.
template <int K, bool FUSE>
__global__ void k_gemm_wmma(const float* __restrict__ A,
                            const float* __restrict__ Agg,
                            const float* __restrict__ invdeg,
                            const float* __restrict__ W,
                            const float* __restrict__ bias,
                            float* __restrict__ C, int M) {
  constexpr int LS = K + 4;                     // padded LDS row stride (floats)
  __shared__ float lds_a[(FUSE ? 2 : 1) * 16 * LS];

  const int row0 = blockIdx.x * 16;
  if (threadIdx.x < 32) {                       // wave 0 issues the DMA
    const unsigned base = (unsigned)(unsigned long long)(void*)lds_a;
    tdm_load_tile16<K>(A + (size_t)row0 * K, base);
    if (FUSE) tdm_load_tile16<K>(Agg + (size_t)row0 * K, base + 16u * LS * 4u);
    __builtin_amdgcn_s_wait_tensorcnt(0);
  }
  __syncthreads();

  const int wave = threadIdx.x >> 5;            // 0..7 -> column tile
  const int lane = threadIdx.x & 31;
  const int col = (wave << 4) + (lane & 15);
  const int mn = lane & 15;                     // row-in-tile
  const int khalf = (lane >> 4) << 1;           // lanes 16-31 hold K+2,K+3
  int ra = row0 + mn;
  if (ra > M - 1) ra = M - 1;
  const float idg = FUSE ? invdeg[ra] : 0.0f;
  const float* __restrict__ hs = lds_a + mn * LS;
  const float* __restrict__ gs = lds_a + (16 + mn) * LS;

  v8f c = {};
#pragma unroll
  for (int k = 0; k < K; k += 4) {
    const int ka = k + khalf;
    v2f a, b;
    a.x = hs[ka];
    a.y = hs[ka + 1];
    if (FUSE) {                                 // compile-time: branch-free loop
      a.x = fmaf(gs[ka], idg, a.x);
      a.y = fmaf(gs[ka + 1], idg, a.y);
    }
    b.x = W[(size_t)ka * HID + col];
    b.y = W[(size_t)(ka + 1) * HID + col];
    c = __builtin_amdgcn_wmma_f32_16x16x4_f32(false, a, false, b,
                                              (short)0, c, false, false);
  }

  const float bcol = bias[col];
  const int rbase = row0 + ((lane >> 4) << 3);  // lanes 16-31 -> rows +8..+15
#pragma unroll
  for (int v = 0; v < 8; ++v) {
    const int r = rbase + v;
    if (r < M) C[(size_t)r * HID + col] = c[v] + bcol;
  }
}

// --------------------------------------------------- edge scatter (1 wave/edge)
__global__ void k_scatter(const float* __restrict__ h, const int* __restrict__ src,
                          const int* __restrict__ dst, float* __restrict__ agg, int nE) {
  const long long gid = (long long)blockIdx.x * blockDim.x + threadIdx.x;
  const int e = (int)(gid >> 5);
  const int lane = (int)(gid & 31);
  if (e >= nE) return;
  const int s = src[e];
  const int d = dst[e];
  const float4 v = *(const float4*)(h + (size_t)s * HID + lane * 4);
  float* p = agg + (size_t)d * HID + lane * 4;
  atomicAdd(p + 0, v.x);
  atomicAdd(p + 1, v.y);
  atomicAdd(p + 2, v.z);
  atomicAdd(p + 3, v.w);
}

// --------------------------------- LayerNorm + ReLU + residual (1 wave/node)
__global__ void k_ln_relu_res(const float* __restrict__ t, float* __restrict__ h,
                              const float* __restrict__ g, const float* __restrict__ b,
                              int N) {
  const long long gid = (long long)blockIdx.x * blockDim.x + threadIdx.x;
  const int node = (int)(gid >> 5);
  const int lane = (int)(gid & 31);
  if (node >= N) return;

  const float4 tv = *(const float4*)(t + (size_t)node * HID + lane * 4);
  float s = tv.x + tv.y + tv.z + tv.w;
#pragma unroll
  for (int off = 16; off > 0; off >>= 1) s += __shfl_xor(s, off, 32);
  const float mu = s * (1.0f / 128.0f);

  const float dx = tv.x - mu, dy = tv.y - mu, dz = tv.z - mu, dw = tv.w - mu;
  float v = dx * dx + dy * dy + dz * dz + dw * dw;
#pragma unroll
  for (int off = 16; off > 0; off >>= 1) v += __shfl_xor(v, off, 32);
  const float rstd = rsqrtf(v * (1.0f / 128.0f) + 1e-5f);

  const float4 res = *(const float4*)(h + (size_t)node * HID + lane * 4);
  const float4 gg  = *(const float4*)(g + lane * 4);
  const float4 bb  = *(const float4*)(b + lane * 4);
  float4 o;
  o.x = fmaxf(fmaf(dx * rstd, gg.x, bb.x), 0.0f) + res.x;
  o.y = fmaxf(fmaf(dy * rstd, gg.y, bb.y), 0.0f) + res.y;
  o.z = fmaxf(fmaf(dz * rstd, gg.z, bb.z), 0.0f) + res.z;
  o.w = fmaxf(fmaf(dw * rstd, gg.w, bb.w), 0.0f) + res.w;
  *(float4*)(h + (size_t)node * HID + lane * 4) = o;
}

// ---------------------------------------- final: mean over rows, then matvec
__global__ void k_colsum(const float* __restrict__ h, float* __restrict__ acc,
                         const int* __restrict__ na_ptr) {
  const int c = threadIdx.x;              // 128 threads: one per column
  const int na = *na_ptr;
  float s = 0.0f;
  for (int r = blockIdx.x; r < na; r += gridDim.x) s += h[(size_t)r * HID + c];
  atomicAdd(&acc[c], s);
}

__global__ void k_out(const float* __restrict__ acc, const int* __restrict__ na_ptr,
                      const float* __restrict__ Wout, const float* __restrict__ bout,
                      float* __restrict__ out) {
  const int o = threadIdx.x;              // 128 threads
  const float invn = 1.0f / (float)(*na_ptr);
  float s = 0.0f;
#pragma unroll 8
  for (int c = 0; c < HID; ++c) s = fmaf(acc[c] * invn, Wout[c * HID + o], s);
  out[o] = s + bout[o];
}

// ---------------------------------------------------------------- dispatcher
extern "C" void kernel_launch(void* const* d_in, const int* in_sizes, int n_in,
                              void* d_out, int out_size, void* d_ws, size_t ws_size,
                              hipStream_t stream) {
  const float* x        = (const float*)d_in[0];
  const int*   ei       = (const int*)  d_in[1];
  const int*   na_ptr   = (const int*)  d_in[2];
  const float* W_in     = (const float*)d_in[3];
  const float* b_in     = (const float*)d_in[4];
  const float* W_layers = (const float*)d_in[5];
  const float* b_layers = (const float*)d_in[6];
  const float* ln_g     = (const float*)d_in[7];
  const float* ln_b     = (const float*)d_in[8];
  const float* W_out    = (const float*)d_in[9];
  const float* b_out    = (const float*)d_in[10];

  const int N  = in_sizes[0] / 64;       // 50000
  const int nE = in_sizes[1] / 2;        // 800000
  const int* src = ei;
  const int* dst = ei + nE;

  float* ws = (float*)d_ws;
  const size_t o_h   = ((size_t)N + 63) / 64 * 64;   // after inv_deg[N]
  const size_t o_agg = o_h   + (size_t)N * HID;
  const size_t o_tmp = o_agg + (size_t)N * HID;
  const size_t o_acc = o_tmp + (size_t)N * HID;
  float* inv_deg = ws;
  float* h   = ws + o_h;
  float* agg = ws + o_agg;
  float* tmp = ws + o_tmp;
  float* acc = ws + o_acc;

  // degree -> inv_deg (in place)
  k_zero<<<512, 256, 0, stream>>>(inv_deg, (long long)N);
  k_degree<<<(nE + 255) / 256, 256, 0, stream>>>(dst, inv_deg, nE);
  k_invdeg<<<(N + 255) / 256, 256, 0, stream>>>(inv_deg, N);

  // h = x @ W_in + b_in
  const int gblocks = (N + 15) / 16;
  k_gemm_wmma<64, false><<<gblocks, 256, 0, stream>>>(x, nullptr, nullptr,
                                                      W_in, b_in, h, N);

  for (int i = 0; i < 3; ++i) {
    k_zero<<<2048, 256, 0, stream>>>(agg, (long long)N * HID);
    const long long sthreads = (long long)nE * 32;
    k_scatter<<<(unsigned)((sthreads + 255) / 256), 256, 0, stream>>>(h, src, dst, agg, nE);
    k_gemm_wmma<128, true><<<gblocks, 256, 0, stream>>>(h, agg, inv_deg,
                                                        W_layers + (size_t)i * HID * HID,
                                                        b_layers + (size_t)i * HID,
                                                        tmp, N);
    const long long lthreads = (long long)N * 32;
    k_ln_relu_res<<<(unsigned)((lthreads + 255) / 256), 256, 0, stream>>>(
        tmp, h, ln_g + (size_t)i * HID, ln_b + (size_t)i * HID, N);
  }

  // out = mean(h[:na]) @ W_out + b_out   (mean commutes with the matmul)
  k_zero<<<1, 128, 0, stream>>>(acc, HID);
  k_colsum<<<512, 128, 0, stream>>>(h, acc, na_ptr);
  k_out<<<1, 128, 0, stream>>>(acc, na_ptr, W_out, b_out, (float*)d_out);
}